// AttentionHead_40870908788988
// MI455X (gfx1250) — compile-verified
//
#include <hip/hip_runtime.h>
#include <math.h>

// ---------------------------------------------------------------------------
// AttentionHead on MI455X (gfx1250).
//  - fp32 GEMMs on the matrix pipe: V_WMMA_F32_16X16X4_F32 (exact fp32).
//  - global->LDS staging via gfx1250 async LDS DMA (GLOBAL_LOAD_ASYNC_TO_LDS_B128,
//    ASYNCcnt) with double-buffered LDS: next tile's DMA overlaps current WMMAs.
//
// d_out layout (tuple return): [ out | Q | K | V ], each [B=4, d=1024, n=2048].
// d_ws: scores/attn buffer [B, n, n] fp32 = 67.1 MB.
// ---------------------------------------------------------------------------

typedef float v2f __attribute__((ext_vector_type(2)));
typedef float v8f __attribute__((ext_vector_type(8)));

#define BM 128
#define BN 128
#define BK 16
#define APITCH 20            // dword pitch for row-major A tile: 16B-aligned rows,
                             // and (20*m + k) mod 64 distinct for m=0..15 -> no bank conflicts
#define NEG_BIG -1.0e12f

__device__ __forceinline__ void async_copy_b128(const float* gaddr, const void* lds_dst) {
    // GLOBAL_LOAD_ASYNC_TO_LDS_B128 (GV mode): VDST = LDS byte address (low 32
    // bits of the flat pointer per ISA 10.2), VADDR = 64-bit global address.
    unsigned lds_off = (unsigned)(unsigned long long)lds_dst;
    asm volatile("global_load_async_to_lds_b128 %0, %1, off"
                 :: "v"(lds_off), "v"(gaddr)
                 : "memory");
}
__device__ __forceinline__ void wait_async_le4() {
    asm volatile("s_wait_asynccnt 0x4" ::: "memory");
}
__device__ __forceinline__ void wait_async_0() {
    asm volatile("s_wait_asynccnt 0x0" ::: "memory");
}

// C[M,N] = alpha * A x B  (per-batch strides; A optionally accessed transposed)
//   TRANS_A=0: A[m,k] = Ag[m*lda + k]
//   TRANS_A=1: A[m,k] = Ag[k*lda + m]      (used for Q^T)
//   B[k,n] = Bg[k*ldb + n]
// MASK: C[row,col] = NEG_BIG when row>=mask_from && col>=mask_from (scores).
template <bool TRANS_A, bool MASK>
__global__ __launch_bounds__(256) void gemm_f32_wmma(
    const float* __restrict__ Ag, const float* __restrict__ Bg,
    float* __restrict__ Cg, int M, int N, int K, int lda, int ldb, int ldc,
    long long sA, long long sB, long long sC, float alpha,
    const int* __restrict__ mask_from_p)
{
    const int bz = blockIdx.z;
    const float* A = Ag + (long long)bz * sA;
    const float* B = Bg + (long long)bz * sB;
    float*       C = Cg + (long long)bz * sC;

    // Double-buffered LDS tiles, layouts chosen so async b128 copies are
    // contiguous per global row:
    //   TRANS_A=0: Asm[buf][m*APITCH + k]   (row-major, padded pitch)
    //   TRANS_A=1: Asm[buf][k*BM + m]       (k-major; Q rows are contiguous in m)
    //   Bsm[buf][k*BN + n]                  (row-major)
    __shared__ float Asm[2][TRANS_A ? (BK * BM) : (BM * APITCH)];
    __shared__ float Bsm[2][BK * BN];

    const int tid   = threadIdx.x;
    const int lane  = tid & 31;
    const int wave  = tid >> 5;        // 8 waves: 2 along M (64 rows), 4 along N (32 cols)
    const int waveM = (wave & 1) * 64;
    const int waveN = (wave >> 1) * 32;
    const int lhalf = lane >> 4;       // selects K {0,1} vs {2,3} / C rows 0-7 vs 8-15
    const int l15   = lane & 15;

    const int rowBase = blockIdx.x * BM;
    const int colBase = blockIdx.y * BN;

    // ---- async stagers: 2 b128 pieces per thread for A, 2 for B ------------
    auto stageA = [&](int k0, int buf) {
#pragma unroll
        for (int j = 0; j < 2; ++j) {
            int p = tid + j * 256;                 // 0..511
            if (TRANS_A) {
                int e = p >> 5, piece = p & 31;    // 16 rows x 32 pieces (512B rows)
                async_copy_b128(&A[(long long)(k0 + e) * lda + rowBase + piece * 4],
                                &Asm[buf][e * BM + piece * 4]);
            } else {
                int m = p >> 2, piece = p & 3;     // 128 rows x 4 pieces (64B rows)
                async_copy_b128(&A[(long long)(rowBase + m) * lda + k0 + piece * 4],
                                &Asm[buf][m * APITCH + piece * 4]);
            }
        }
    };
    auto stageB = [&](int k0, int buf) {
#pragma unroll
        for (int j = 0; j < 2; ++j) {
            int p = tid + j * 256;                 // 16 rows x 32 pieces (512B rows)
            int krow = p >> 5, piece = p & 31;
            async_copy_b128(&B[(long long)(k0 + krow) * ldb + colBase + piece * 4],
                            &Bsm[buf][krow * BN + piece * 4]);
        }
    };

    v8f acc[4][2];
#pragma unroll
    for (int mi = 0; mi < 4; ++mi)
#pragma unroll
        for (int ni = 0; ni < 2; ++ni)
#pragma unroll
            for (int e = 0; e < 8; ++e) acc[mi][ni][e] = 0.0f;

    // ---- software pipeline: prime buffer 0 --------------------------------
    stageA(0, 0);
    stageB(0, 0);                                  // 4 async instrs in flight

    int buf = 0;
    for (int k0 = 0; k0 < K; k0 += BK) {
        const int knext = k0 + BK;
        if (knext < K) {                           // issue next tile's DMA first
            stageA(knext, buf ^ 1);
            stageB(knext, buf ^ 1);
            wait_async_le4();                      // current tile landed; next in flight
        } else {
            wait_async_0();
        }
        __syncthreads();                           // all waves' DMA visible

        const float* __restrict__ As = Asm[buf];
        const float* __restrict__ Bs = Bsm[buf];
#pragma unroll
        for (int kk = 0; kk < BK; kk += 4) {
            const int kbase = kk + 2 * lhalf;      // K pair this lane-half consumes
            v2f afrag[4], bfrag[2];
#pragma unroll
            for (int mi = 0; mi < 4; ++mi) {
                const int m = waveM + mi * 16 + l15;
                if (TRANS_A) {
                    afrag[mi] = v2f{As[kbase * BM + m], As[(kbase + 1) * BM + m]};
                } else {
                    afrag[mi] = *reinterpret_cast<const v2f*>(&As[m * APITCH + kbase]);
                }
            }
#pragma unroll
            for (int ni = 0; ni < 2; ++ni) {
                const int nn = waveN + ni * 16 + l15;
                bfrag[ni] = v2f{Bs[kbase * BN + nn], Bs[(kbase + 1) * BN + nn]};
            }
#pragma unroll
            for (int mi = 0; mi < 4; ++mi)
#pragma unroll
                for (int ni = 0; ni < 2; ++ni)
                    acc[mi][ni] = __builtin_amdgcn_wmma_f32_16x16x4_f32(
                        false, afrag[mi], false, bfrag[ni],
                        (short)0, acc[mi][ni], false, false);
        }
        __syncthreads();                           // done reading buf -> reusable
        buf ^= 1;
    }

    // ---- epilogue: C layout = VGPR r, lane half -> M = r + 8*half, N = l15
    const int mf = MASK ? *mask_from_p : 0;
#pragma unroll
    for (int mi = 0; mi < 4; ++mi)
#pragma unroll
        for (int ni = 0; ni < 2; ++ni)
#pragma unroll
            for (int r = 0; r < 8; ++r) {
                int row = rowBase + waveM + mi * 16 + r + 8 * lhalf;
                int col = colBase + waveN + ni * 16 + l15;
                float v = alpha * acc[mi][ni][r];
                if (MASK && row >= mf && col >= mf) v = NEG_BIG;
                C[(long long)row * ldc + col] = v;
            }
}

// softmax over the BATCH axis (legacy torch implicit dim=0 on 3D): 4-way, in place.
__global__ __launch_bounds__(256) void softmax_batch4(float* __restrict__ S,
                                                      long long plane)
{
    long long idx = (long long)blockIdx.x * 256 + threadIdx.x;
    if (idx >= plane) return;
    float s0 = S[idx];
    float s1 = S[idx + plane];
    float s2 = S[idx + 2 * plane];
    float s3 = S[idx + 3 * plane];
    float m  = fmaxf(fmaxf(s0, s1), fmaxf(s2, s3));
    float e0 = __expf(s0 - m), e1 = __expf(s1 - m);
    float e2 = __expf(s2 - m), e3 = __expf(s3 - m);
    float inv = 1.0f / (e0 + e1 + e2 + e3);
    S[idx]             = e0 * inv;
    S[idx + plane]     = e1 * inv;
    S[idx + 2 * plane] = e2 * inv;
    S[idx + 3 * plane] = e3 * inv;
}

extern "C" void kernel_launch(void* const* d_in, const int* in_sizes, int n_in,
                              void* d_out, int out_size, void* d_ws, size_t ws_size,
                              hipStream_t stream)
{
    const float* x  = (const float*)d_in[0];   // [B, d, n]
    const float* Wq = (const float*)d_in[1];   // [d, d]
    const float* Wk = (const float*)d_in[2];
    const float* Wv = (const float*)d_in[3];
    const int* mask_from = (const int*)d_in[4];

    const int B = 4, d = 1024, n = 2048;
    const long long dn = (long long)d * n;     // 2,097,152
    const long long nn = (long long)n * n;     // 4,194,304

    float* out = (float*)d_out;                // tuple: (out, Q, K, V)
    float* Q   = out + (long long)B * dn;
    float* Kt  = Q   + (long long)B * dn;
    float* V   = Kt  + (long long)B * dn;
    float* S   = (float*)d_ws;                 // [B, n, n] scores/attn (67.1 MB)

    dim3 blk(256);

    // Q/K/V = W @ x  :  [d,n] = [d,d] x [d,n], batched over B (W stride 0)
    dim3 gqkv(d / BM, n / BN, B);              // (8,16,4)
    gemm_f32_wmma<false, false><<<gqkv, blk, 0, stream>>>(
        Wq, x, Q,  d, n, d, d, n, n, 0LL, dn, dn, 1.0f, nullptr);
    gemm_f32_wmma<false, false><<<gqkv, blk, 0, stream>>>(
        Wk, x, Kt, d, n, d, d, n, n, 0LL, dn, dn, 1.0f, nullptr);
    gemm_f32_wmma<false, false><<<gqkv, blk, 0, stream>>>(
        Wv, x, V,  d, n, d, d, n, n, 0LL, dn, dn, 1.0f, nullptr);

    // scores = Q^T K / 32, masked quadrant -> -1e12  :  [n,n], batched
    dim3 gs(n / BM, n / BN, B);                // (16,16,4)
    gemm_f32_wmma<true, true><<<gs, blk, 0, stream>>>(
        Q, Kt, S, n, n, d, n, n, n, dn, dn, nn, 0.03125f, mask_from);

    // attn = softmax over batch axis (4-way), in place
    softmax_batch4<<<(unsigned)((nn + 255) / 256), blk, 0, stream>>>(S, nn);

    // out = V @ attn  :  [d,n] = [d,n] x [n,n], batched
    dim3 go(d / BM, n / BN, B);                // (8,16,4)
    gemm_f32_wmma<false, false><<<go, blk, 0, stream>>>(
        V, S, out, d, n, n, n, n, n, dn, nn, dn, 1.0f, nullptr);
}